// RIOT_5076651344440
// MI455X (gfx1250) — compile-verified
//
#include <hip/hip_runtime.h>
#include <math.h>

typedef float v2f __attribute__((ext_vector_type(2)));
typedef float v4f __attribute__((ext_vector_type(4)));
typedef float v8f __attribute__((ext_vector_type(8)));

#define NN 4096
#define NN4 1024            // NN/4, row stride in v4f units
#define PD 128

// ---------------- T = U @ A  (4096x128, tiny) ----------------
__global__ __launch_bounds__(256) void k_ua(const float* __restrict__ U,
                                            const float* __restrict__ A,
                                            float* __restrict__ T) {
    int idx = blockIdx.x * 256 + threadIdx.x;        // 4096*128 threads
    int i = idx >> 7;
    int q = idx & 127;
    const float* u = U + (size_t)i * PD;
    float s = 0.f;
#pragma unroll 8
    for (int p = 0; p < PD; ++p) s = fmaf(u[p], A[p * PD + q], s);
    T[idx] = s;
}

// ------- K = exp(-softplus(W - T V^T)), 16x16 tiles via V_WMMA_F32_16X16X4_F32 -------
// One wave32 per 16x16 tile. ISA 7.12.2 layout (32-bit A 16x4):
// lane<16 -> K={k,k+1}, lane>=16 -> K={k+2,k+3}, M=lane&15. B = V^T gives the
// identical addressing pattern into V. C/D: row = r + 8*(lane>=16), col = lane&15.
__global__ __launch_bounds__(256) void k_K(const float* __restrict__ T,
                                           const float* __restrict__ V,
                                           const float* __restrict__ W,
                                           float* __restrict__ K) {
    int lane = threadIdx.x;                 // 0..31
    int wave = threadIdx.y;                 // 0..7
    int rowTile = blockIdx.y * 8 + wave;    // 0..255
    int colTile = blockIdx.x;               // 0..255
    int m0 = rowTile * 16;
    int n0 = colTile * 16;
    int m  = lane & 15;
    int hi = lane >> 4;                     // 0 or 1

    const float* trow = T + (size_t)(m0 + m) * PD + 2 * hi;
    const float* vrow = V + (size_t)(n0 + m) * PD + 2 * hi;

    v8f c = {};
#pragma unroll 4
    for (int k = 0; k < PD; k += 4) {
        v2f af, bf;
        af.x = trow[k];  af.y = trow[k + 1];
        bf.x = vrow[k];  bf.y = vrow[k + 1];
        c = __builtin_amdgcn_wmma_f32_16x16x4_f32(
                /*neg_a=*/false, af, /*neg_b=*/false, bf,
                /*c_mod=*/(short)0, c, /*reuse_a=*/false, /*reuse_b=*/false);
    }

    int col = n0 + m;
#pragma unroll
    for (int r = 0; r < 8; ++r) {
        int row = m0 + r + 8 * hi;
        size_t idx = (size_t)row * NN + col;
        float w = __builtin_nontemporal_load(&W[idx]);   // W is read-once: don't cache
        float x = w - c[r];
        // stable softplus
        float sp = (x > 0.f) ? (x + log1pf(expf(-x))) : log1pf(expf(x));
        K[idx] = expf(-sp);                 // LAMBDA = 1; K stays L2-resident
    }
}

// ---------------- b init to ones ----------------
__global__ __launch_bounds__(256) void k_init(float* __restrict__ b) {
    b[blockIdx.x * 256 + threadIdx.x] = 1.f;
}

// ---------------- a = mu / (K @ b): one block per row, 16B loads, LDS tree reduce ----------------
__global__ __launch_bounds__(256) void k_row(const float* __restrict__ K,
                                             const float* __restrict__ b,
                                             const float* __restrict__ mu,
                                             float* __restrict__ a) {
    __shared__ float red[256];
    int row = blockIdx.x;
    const v4f* kr = (const v4f*)(K + (size_t)row * NN);
    const v4f* b4 = (const v4f*)b;
    float s = 0.f;
#pragma unroll
    for (int j = threadIdx.x; j < NN4; j += 256) {
        v4f kv = kr[j];
        v4f bv = b4[j];
        s = fmaf(kv.x, bv.x, s);
        s = fmaf(kv.y, bv.y, s);
        s = fmaf(kv.z, bv.z, s);
        s = fmaf(kv.w, bv.w, s);
    }
    red[threadIdx.x] = s;
    __syncthreads();
#pragma unroll
    for (int off = 128; off > 0; off >>= 1) {
        if (threadIdx.x < off) red[threadIdx.x] += red[threadIdx.x + off];
        __syncthreads();
    }
    if (threadIdx.x == 0) a[row] = mu[row] / red[0];
}

// ---- partial[chunk][:] = sum over 128-row chunk of K[i][col]*a[i], 4 cols/thread ----
__global__ __launch_bounds__(256) void k_colp(const float* __restrict__ K,
                                              const float* __restrict__ a,
                                              float* __restrict__ partial) {
    int c4 = blockIdx.x * 256 + threadIdx.x;    // 0..1023 (grid.x = 4)
    int i0 = blockIdx.y * 128;                  // 32 chunks in y
    const v4f* K4 = (const v4f*)K;
    v4f s = {0.f, 0.f, 0.f, 0.f};
#pragma unroll 4
    for (int i = i0; i < i0 + 128; ++i) {
        v4f kv = K4[(size_t)i * NN4 + c4];
        float av = a[i];
        s.x = fmaf(kv.x, av, s.x);
        s.y = fmaf(kv.y, av, s.y);
        s.z = fmaf(kv.z, av, s.z);
        s.w = fmaf(kv.w, av, s.w);
    }
    ((v4f*)partial)[(size_t)blockIdx.y * NN4 + c4] = s;
}

// ---------------- b = nu / sum_chunks(partial) ----------------
__global__ __launch_bounds__(256) void k_colr(const float* __restrict__ partial,
                                              const float* __restrict__ nu,
                                              float* __restrict__ b) {
    int c4 = blockIdx.x * 256 + threadIdx.x;    // grid.x = 4
    const v4f* p4 = (const v4f*)partial;
    v4f s = {0.f, 0.f, 0.f, 0.f};
#pragma unroll
    for (int c = 0; c < 32; ++c) {
        v4f pv = p4[(size_t)c * NN4 + c4];
        s.x += pv.x; s.y += pv.y; s.z += pv.z; s.w += pv.w;
    }
    v4f nv = ((const v4f*)nu)[c4];
    v4f r;
    r.x = nv.x / s.x; r.y = nv.y / s.y; r.z = nv.z / s.z; r.w = nv.w / s.w;
    ((v4f*)b)[c4] = r;
}

// ---------------- out = a * K * b^T (NT store: don't evict K from L2) ----------------
__global__ __launch_bounds__(256) void k_plan(const float* __restrict__ K,
                                              const float* __restrict__ a,
                                              const float* __restrict__ b,
                                              float* __restrict__ out) {
    int j4 = blockIdx.x * 256 + threadIdx.x;    // grid (4, 4096)
    int i  = blockIdx.y;
    size_t idx = (size_t)i * NN4 + j4;
    v4f kv = ((const v4f*)K)[idx];
    v4f bv = ((const v4f*)b)[j4];
    float ai = a[i];
    v4f o;
    o.x = ai * kv.x * bv.x;
    o.y = ai * kv.y * bv.y;
    o.z = ai * kv.z * bv.z;
    o.w = ai * kv.w * bv.w;
    __builtin_nontemporal_store(o, &((v4f*)out)[idx]);
}

extern "C" void kernel_launch(void* const* d_in, const int* in_sizes, int n_in,
                              void* d_out, int out_size, void* d_ws, size_t ws_size,
                              hipStream_t stream) {
    const float* U  = (const float*)d_in[0];
    const float* V  = (const float*)d_in[1];
    const float* W  = (const float*)d_in[2];
    const float* mu = (const float*)d_in[3];
    const float* nu = (const float*)d_in[4];
    const float* A  = (const float*)d_in[5];
    float* out = (float*)d_out;

    char* ws = (char*)d_ws;
    size_t offT = 0;
    size_t offK = offT + (size_t)NN * PD * sizeof(float);          // 2 MB
    size_t offA = offK + (size_t)NN * NN * sizeof(float);          // +64 MB
    size_t offB = offA + (size_t)NN * sizeof(float);
    size_t offP = offB + (size_t)NN * sizeof(float);
    float* T       = (float*)(ws + offT);
    float* K       = (float*)(ws + offK);
    float* a_vec   = (float*)(ws + offA);
    float* b_vec   = (float*)(ws + offB);
    float* partial = (float*)(ws + offP);                          // 32*4096 floats

    // Phase 1: T = U @ A
    k_ua<<<(NN * PD) / 256, 256, 0, stream>>>(U, A, T);

    // Phase 2: K via fp32 WMMA tiles; K becomes L2-resident (64MB < 192MB L2)
    k_K<<<dim3(NN / 16, NN / 128), dim3(32, 8), 0, stream>>>(T, V, W, K);

    // Phase 3: Sinkhorn, 100 iterations; fixed-order reductions => deterministic
    k_init<<<NN / 256, 256, 0, stream>>>(b_vec);
    for (int it = 0; it < 100; ++it) {
        k_row<<<NN, 256, 0, stream>>>(K, b_vec, mu, a_vec);
        k_colp<<<dim3(NN4 / 256, 32), 256, 0, stream>>>(K, a_vec, partial);
        k_colr<<<NN4 / 256, 256, 0, stream>>>(partial, nu, b_vec);
    }

    // Phase 4: plan = a * K * b^T
    k_plan<<<dim3(NN4 / 256, NN), 256, 0, stream>>>(K, a_vec, b_vec, out);
}